// SoftAdvectSparseConservative_84585085928010
// MI455X (gfx1250) — compile-verified
//
#include <hip/hip_runtime.h>
#include <math.h>

#define BN 2
#define XN 512
#define YN 512
#define TN 4
#define NN (BN * XN * YN * TN)   // 2097152
#define CN 32
#define MAXS 2.5f

typedef __attribute__((ext_vector_type(2))) float v2f;
typedef __attribute__((ext_vector_type(4))) float v4f;
typedef __attribute__((ext_vector_type(8))) float v8f;
typedef __attribute__((ext_vector_type(4))) int   v4i;

#define AS1 __attribute__((address_space(1)))
#define AS3 __attribute__((address_space(3)))

#if __has_builtin(__builtin_amdgcn_global_load_async_to_lds_b128)
#define HAVE_ASYNC_LDS 1
#else
#define HAVE_ASYNC_LDS 0
#endif

__device__ __forceinline__ int lin_idx(int b, int x, int y, int t) {
  return ((b * XN + x) * YN + y) * TN + t;
}

struct Tgt { int x0, y0, t1, b; float w[4]; };

// Per-source splat geometry: clipped velocity, floor cell, 4 Gaussian weights.
__device__ __forceinline__ Tgt calc_tgt(const int* __restrict__ coords,
                                        const float* __restrict__ vel, int n) {
  Tgt r;
  const int b = coords[4 * n + 0];
  const int x = coords[4 * n + 1];
  const int y = coords[4 * n + 2];
  const int t = coords[4 * n + 3];
  const float vx = fminf(fmaxf(vel[2 * n + 0], -MAXS), MAXS);
  const float vy = fminf(fmaxf(vel[2 * n + 1], -MAXS), MAXS);
  const float xt = (float)x + vx;
  const float yt = (float)y + vy;
  const float x0f = floorf(xt), y0f = floorf(yt);
  r.x0 = (int)x0f; r.y0 = (int)y0f; r.t1 = t + 1; r.b = b;
  const float dx0 = fabsf(xt - x0f), dx1 = fabsf(xt - (x0f + 1.0f));
  const float dy0 = fabsf(yt - y0f), dy1 = fabsf(yt - (y0f + 1.0f));
  r.w[0] = expf(-(dx0 * dx0 + dy0 * dy0));
  r.w[1] = expf(-(dx1 * dx1 + dy0 * dy0));
  r.w[2] = expf(-(dx0 * dx0 + dy1 * dy1));
  r.w[3] = expf(-(dx1 * dx1 + dy1 * dy1));
  return r;
}

// Pass 1: perfect-hash map (coords is a dense permutation -> every slot written).
__global__ __launch_bounds__(256) void k_map(const int* __restrict__ coords,
                                             int* __restrict__ map) {
  const int n = blockIdx.x * 256 + threadIdx.x;
  const int b = coords[4 * n + 0];
  const int x = coords[4 * n + 1];
  const int y = coords[4 * n + 2];
  const int t = coords[4 * n + 3];
  map[lin_idx(b, x, y, t)] = n;
}

// Pass 2: destination weight totals (atomics into 8MB L2-resident array).
__global__ __launch_bounds__(256) void k_wsum(const int* __restrict__ coords,
                                              const float* __restrict__ vel,
                                              const int* __restrict__ map,
                                              float* __restrict__ wsum) {
  const int n = blockIdx.x * 256 + threadIdx.x;
  const Tgt t = calc_tgt(coords, vel, n);
  if (t.t1 < TN) {
#pragma unroll
    for (int k = 0; k < 4; ++k) {
      const int nx = t.x0 + (k & 1);
      const int ny = t.y0 + (k >> 1);
      if (nx >= 0 && nx < XN && ny >= 0 && ny < YN) {
        atomicAdd(&wsum[map[lin_idx(t.b, nx, ny, t.t1)]], t.w[k]);
      }
    }
  }
}

// Pass 3: normalized feature scatter. One wave32 per source row: the 32
// features are one coalesced 128B read and 4 lane-parallel f32 atomic bursts.
__global__ __launch_bounds__(256) void k_scatter(const int* __restrict__ coords,
                                                 const float* __restrict__ vel,
                                                 const int* __restrict__ map,
                                                 const float* __restrict__ wsum,
                                                 const float* __restrict__ feats,
                                                 float* __restrict__ accum) {
  const int n = blockIdx.x * 8 + (threadIdx.x >> 5);
  const int lane = threadIdx.x & 31;
  const Tgt t = calc_tgt(coords, vel, n);   // wave-uniform
  const float f = feats[(size_t)n * CN + lane];
  if (t.t1 < TN) {
#pragma unroll
    for (int k = 0; k < 4; ++k) {
      const int nx = t.x0 + (k & 1);
      const int ny = t.y0 + (k >> 1);
      if (nx >= 0 && nx < XN && ny >= 0 && ny < YN) {
        const int dst = map[lin_idx(t.b, nx, ny, t.t1)];
        const float nw = t.w[k] / fmaxf(wsum[dst], 1e-6f);
        atomicAdd(&accum[(size_t)dst * CN + lane], nw * f);  // GAIN == 1.0
      }
    }
  }
}

// Pass 4: gate. 16 rows per wave. Tiles staged via async global->LDS copy,
// both 32-wide L1 row reductions done as f32 WMMA 16x16x4 chains (B = ones).
__global__ __launch_bounds__(256) void k_gate(const float* __restrict__ feats,
                                              const float* __restrict__ vel,
                                              float* out) {
  __shared__ float sF[8][16 * CN];
  __shared__ float sA[8][16 * CN];
  const int wv = threadIdx.x >> 5;
  const int l  = threadIdx.x & 31;
  const int r0 = (blockIdx.x * 8 + wv) * 16;

  const float* gF = feats + (size_t)r0 * CN;
  const float* gA = out   + (size_t)r0 * CN;   // accum (read phase)

#if HAVE_ASYNC_LDS
#pragma unroll
  for (int i = 0; i < 4; ++i) {
    const int e = i * 128 + l * 4;
    __builtin_amdgcn_global_load_async_to_lds_b128(
        (AS1 v4i*)(uintptr_t)(gF + e), (AS3 v4i*)(uintptr_t)(&sF[wv][e]), 0, 0);
    __builtin_amdgcn_global_load_async_to_lds_b128(
        (AS1 v4i*)(uintptr_t)(gA + e), (AS3 v4i*)(uintptr_t)(&sA[wv][e]), 0, 0);
  }
#if __has_builtin(__builtin_amdgcn_s_wait_asynccnt)
  __builtin_amdgcn_s_wait_asynccnt(0);
#else
  asm volatile("s_wait_asynccnt 0" ::: "memory");
#endif
  asm volatile("" ::: "memory");
#else
#pragma unroll
  for (int i = 0; i < 4; ++i) {
    const int e = i * 128 + l * 4;
    *(v4f*)(&sF[wv][e]) = *(const v4f*)(gF + e);
    *(v4f*)(&sA[wv][e]) = *(const v4f*)(gA + e);
  }
#endif

  const int m = l & 15;   // row within tile (A-matrix M)
  const int h = l >> 4;   // K-half select per 16x4 f32 A layout

  // speed of row r0+m (lane pairs duplicate; broadcast later via shfl)
  float spd;
  {
    const int rr = r0 + m;
    const float vx = fminf(fmaxf(vel[2 * rr + 0], -MAXS), MAXS);
    const float vy = fminf(fmaxf(vel[2 * rr + 1], -MAXS), MAXS);
    spd = fabsf(vx) + fabsf(vy);
  }

  v8f cD = {};            // sum_c |accum - feats| per row
  v8f cL = {};            // sum_c |feats| per row
  const v2f onesB = {1.0f, 1.0f};
  const float* fr = &sF[wv][m * CN];
  const float* ar = &sA[wv][m * CN];
#pragma unroll
  for (int k = 0; k < 8; ++k) {
    // A 16x4 f32 layout: lane m holds K=2h,2h+1 in VGPR0,VGPR1
    const int c0 = 4 * k + 2 * h;
    const float f0 = fr[c0], f1 = fr[c0 + 1];
    const float a0 = ar[c0], a1 = ar[c0 + 1];
    v2f Ad = { fabsf(a0 - f0), fabsf(a1 - f1) };
    v2f Al = { fabsf(f0),      fabsf(f1)      };
    cD = __builtin_amdgcn_wmma_f32_16x16x4_f32(false, Ad, false, onesB,
                                               (short)0, cD, false, false);
    cL = __builtin_amdgcn_wmma_f32_16x16x4_f32(false, Al, false, onesB,
                                               (short)0, cL, false, false);
  }

  // D layout: VGPR i holds row i (lanes 0-15) / row i+8 (lanes 16-31),
  // replicated across the 16 columns since B is all-ones.
#pragma unroll
  for (int r = 0; r < 16; ++r) {
    const int srcLane = (r & 8) ? 16 : 0;
    const float dsum = __shfl(cD[r & 7], srcLane, 32);
    const float lsum = __shfl(cL[r & 7], srcLane, 32);
    const float sp   = __shfl(spd, r, 32);
    const float diff = dsum / fmaxf(lsum, 1e-6f);
    const float gate = expf(-diff) * (1.0f / (1.0f + 0.25f * sp));
    const float f = sF[wv][r * CN + l];
    const float a = sA[wv][r * CN + l];
    out[(size_t)(r0 + r) * CN + l] = gate * a + (1.0f - gate) * f;
  }
}

extern "C" void kernel_launch(void* const* d_in, const int* in_sizes, int n_in,
                              void* d_out, int out_size, void* d_ws, size_t ws_size,
                              hipStream_t stream) {
  (void)in_sizes; (void)n_in; (void)out_size; (void)ws_size;
  const int*   coords = (const int*)d_in[0];
  const float* feats  = (const float*)d_in[1];
  const float* vel    = (const float*)d_in[2];
  float* out = (float*)d_out;

  int*   map  = (int*)d_ws;                                  // 8 MB
  float* wsum = (float*)((char*)d_ws + (size_t)NN * 4);      // 8 MB

  (void)hipMemsetAsync(wsum, 0, (size_t)NN * sizeof(float), stream);
  (void)hipMemsetAsync(out,  0, (size_t)NN * CN * sizeof(float), stream);

  k_map    <<<NN / 256, 256, 0, stream>>>(coords, map);
  k_wsum   <<<NN / 256, 256, 0, stream>>>(coords, vel, map, wsum);
  k_scatter<<<NN / 8,   256, 0, stream>>>(coords, vel, map, wsum, feats, out);
  k_gate   <<<NN / 128, 256, 0, stream>>>(feats, vel, out);
}